// GNN_26912265076901
// MI455X (gfx1250) — compile-verified
//
#include <hip/hip_runtime.h>
#include <hip/hip_bf16.h>

// ---------------------------------------------------------------------------
// CDNA5 (gfx1250) GCN pipeline.  fp32 everywhere (workload is HBM/atomic
// bound, not flop bound); dense per-node GEMMs use V_WMMA_F32_16X16X4_F32
// with branch-free operand masking and wave-uniform store guards.
// ---------------------------------------------------------------------------

typedef __attribute__((ext_vector_type(2))) float v2f;
typedef __attribute__((ext_vector_type(8))) float v8f;

// -------------------------------- utility ----------------------------------

__global__ void zero_f32_kernel(float* __restrict__ p, long long count) {
    long long i = (long long)blockIdx.x * blockDim.x + threadIdx.x;
    long long stride = (long long)gridDim.x * blockDim.x;
    for (; i < count; i += stride) p[i] = 0.0f;
}

// ------------------------------ degree pass --------------------------------

__global__ void deg_kernel(const int* __restrict__ dst, float* __restrict__ degraw, int E) {
    int e = blockIdx.x * blockDim.x + threadIdx.x;
    if (e >= E) return;
    atomicAdd(&degraw[dst[e]], 1.0f);
}

__global__ void dinv_kernel(const float* __restrict__ degraw,
                            float* __restrict__ dinv,
                            float* __restrict__ invdeg, int n) {
    int i = blockIdx.x * blockDim.x + threadIdx.x;
    if (i >= n) return;
    float d = degraw[i] + 1.0f;       // self-loop
    dinv[i]   = rsqrtf(d);            // deg^-1/2
    invdeg[i] = 1.0f / d;             // deg^-1   (self-loop norm)
}

// --------------------------- WMMA fp32 GEMM --------------------------------
// H[n,F] = A[n,K] @ W[K,F].  One wave owns a 16x16 output tile; K marched in
// steps of 4 via v_wmma_f32_16x16x4_f32.  Block = 128 threads = 4 waves.
//
// Operand layouts per CDNA5 ISA 7.12.2:
//   A (16x4):  lane%16 -> M; VGPR0/1 hold K = kb+{0,1} (lanes 0-15) and
//              kb+{2,3} (lanes 16-31).
//   B (4x16):  lane%16 -> N; K rows split low/high lanes symmetrically.
//   D (16x16): VGPR r -> row r (low lanes) / r+8 (high lanes); lane%16 -> N.
//
// Boundary handling is branch-free: each lane's (k, k+1) pair starts at an
// even k (K is even), so the pair is all-valid or all-invalid.  We clamp the
// offset (v_cndmask), load unconditionally as float2, and scale by a 0/1
// mask.  Stores: only the last m-tile can be partial, so the bounds branch
// is wave-uniform; the hot path is 8 unguarded global_store_b32 with
// immediate offsets off one precomputed row base.

template<int K, int F>
__global__ void gemm_wmma_kernel(const float* __restrict__ A,
                                 const float* __restrict__ W,
                                 float* __restrict__ H, int n) {
    constexpr int NT    = F / 16;
    constexpr int STEPS = (K + 3) / 4;
    static_assert((K & 1) == 0, "K must be even for paired loads");

    int wave = threadIdx.x >> 5;
    int lane = threadIdx.x & 31;
    int tile = blockIdx.x * 4 + wave;

    int mtiles = (n + 15) / 16;
    if (tile >= mtiles * NT) return;          // wave-uniform exit

    int mt = tile / NT;
    int nt = tile % NT;
    bool hi  = (lane & 16) != 0;
    int  l16 = lane & 15;

    int m    = mt * 16 + l16;                 // A row for this lane
    int ncol = nt * 16 + l16;                 // B/D column for this lane
    int mload = m < n ? m : (n - 1);          // clamped row for loads

    const float* __restrict__ Arow = A + (long long)mload * K;
    const float* __restrict__ Wcol = W + ncol;

    v8f c = {};
#pragma unroll
    for (int s = 0; s < STEPS; ++s) {
        int  k0  = s * 4 + (hi ? 2 : 0);      // even; pair is (k0, k0+1)
        bool pv  = (k0 + 1) < K;              // pair-valid (uniform per pair)
        int  kc  = pv ? k0 : 0;               // clamped, always in range
        float mk = pv ? 1.0f : 0.0f;

        v2f av = *(const v2f*)(Arow + kc);    // global_load_b64, 8B aligned
        v2f a, b;
        a.x = av.x * mk;
        a.y = av.y * mk;
        b.x = Wcol[kc * F] * mk;
        b.y = Wcol[(kc + 1) * F] * mk;

        // 8 args: (neg_a, A, neg_b, B, c_mod, C, reuse_a, reuse_b)
        c = __builtin_amdgcn_wmma_f32_16x16x4_f32(false, a, false, b,
                                                  (short)0, c, false, false);
    }

    // Row base for this lane's D entries: rows mt*16+(hi?8:0)+r, col ncol.
    int mrow0 = mt * 16 + (hi ? 8 : 0);
    float* __restrict__ Hrow = H + (long long)mrow0 * F + ncol;

    if (mt * 16 + 16 <= n) {                  // wave-uniform: full tile
#pragma unroll
        for (int r = 0; r < 8; ++r)
            Hrow[(long long)r * F] = c[r];    // immediate-offset stores
    } else {                                  // partial last tile
#pragma unroll
        for (int r = 0; r < 8; ++r)
            if (mrow0 + r < n) Hrow[(long long)r * F] = c[r];
    }
}

// --------------------------- edge scatter-add ------------------------------
// agg[dst] += h[src] * dinv[src]*dinv[dst], F4 float4's per edge (template
// so index math is shifts, not a runtime divide).  Random-src gathers hit
// the 192MB L2 (h1 is 25MB); scatters are global f32 atomics.

template<int F4>
__global__ void edge_agg_kernel(const float4* __restrict__ h,
                                float* __restrict__ agg,
                                const int* __restrict__ src,
                                const int* __restrict__ dst,
                                const float* __restrict__ dinv,
                                int E) {
    int idx = blockIdx.x * blockDim.x + threadIdx.x;
    int e = idx / F4;                 // constexpr F4 -> shift
    int f = idx % F4;                 // constexpr F4 -> and
    if (e >= E) return;

    int s = src[e];
    int d = dst[e];
    float norm = dinv[s] * dinv[d];

    float4 v = h[(long long)s * F4 + f];
    float* out = agg + ((long long)d * F4 + f) * 4;
    atomicAdd(out + 0, v.x * norm);
    atomicAdd(out + 1, v.y * norm);
    atomicAdd(out + 2, v.z * norm);
    atomicAdd(out + 3, v.w * norm);
}

// -------------------- self-loop + bias + ReLU (in place) -------------------

template<int F>
__global__ void post_kernel(float* __restrict__ agg,
                            const float* __restrict__ h,
                            const float* __restrict__ invdeg,
                            const float* __restrict__ bias,
                            int n) {
    long long i = (long long)blockIdx.x * blockDim.x + threadIdx.x;
    if (i >= (long long)n * F) return;
    int m = (int)(i / F);             // constexpr F -> shift
    int f = (int)(i % F);             // constexpr F -> and
    float v = agg[i] + h[i] * invdeg[m] + bias[f];
    agg[i] = v > 0.0f ? v : 0.0f;
}

// ------------------------------ graph pool ---------------------------------

__global__ void pool_kernel(const float4* __restrict__ h,   // [n, 8] float4
                            const int* __restrict__ batch,
                            float* __restrict__ pooled,     // [G, 32]
                            int n) {
    int i = blockIdx.x * blockDim.x + threadIdx.x;   // n*8 threads
    if (i >= n * 8) return;
    int node = i >> 3;
    int f4   = i & 7;
    int g = batch[node];
    float4 v = h[(long long)node * 8 + f4];
    float* p = pooled + (long long)g * 32 + f4 * 4;
    atomicAdd(p + 0, v.x);
    atomicAdd(p + 1, v.y);
    atomicAdd(p + 2, v.z);
    atomicAdd(p + 3, v.w);
}

// ------------------------------ FC + sigmoid -------------------------------

__global__ void fc_kernel(const float* __restrict__ pooled,
                          const float* __restrict__ Wfc,
                          const float* __restrict__ bfc,
                          float* __restrict__ out, int G) {
    int g = blockIdx.x * blockDim.x + threadIdx.x;
    if (g >= G) return;
    float z = bfc[0];
#pragma unroll
    for (int k = 0; k < 32; ++k) z += pooled[(long long)g * 32 + k] * Wfc[k];
    out[g] = 1.0f / (1.0f + expf(-z));
}

// ------------------------------- launcher ----------------------------------

extern "C" void kernel_launch(void* const* d_in, const int* in_sizes, int n_in,
                              void* d_out, int out_size, void* d_ws, size_t ws_size,
                              hipStream_t stream) {
    const float* x    = (const float*)d_in[0];     // [n,10]
    const int*   ei   = (const int*)d_in[1];       // [2,E]
    const int*   batch= (const int*)d_in[2];       // [n]
    const float* W1   = (const float*)d_in[3];     // [10,64]
    const float* b1   = (const float*)d_in[4];     // [64]
    const float* W2   = (const float*)d_in[5];     // [64,32]
    const float* b2   = (const float*)d_in[6];     // [32]
    const float* Wfc  = (const float*)d_in[7];     // [32,1]
    const float* bfc  = (const float*)d_in[8];     // [1]
    float* out = (float*)d_out;

    const int n = in_sizes[2];          // N_NODES
    const int E = in_sizes[1] / 2;      // N_EDGES
    const int G = out_size;             // N_GRAPHS

    const int* src = ei;
    const int* dst = ei + E;

    // workspace layout (floats)
    float* ws     = (float*)d_ws;
    float* h1     = ws;                         // n*64
    float* h2     = h1 + (long long)n * 64;     // n*32
    float* dinv   = h2 + (long long)n * 32;     // n
    float* invdeg = dinv + n;                   // n
    // contiguous zero-init region:
    float* degraw = invdeg + n;                 // n
    float* agg1   = degraw + n;                 // n*64
    float* agg2   = agg1 + (long long)n * 64;   // n*32
    float* pooled = agg2 + (long long)n * 32;   // G*32
    long long zcount = (long long)n * 97 + (long long)G * 32;

    // 1. zero deg/agg1/agg2/pooled
    zero_f32_kernel<<<2048, 256, 0, stream>>>(degraw, zcount);

    // 2. degrees -> dinv, invdeg
    deg_kernel<<<(E + 255) / 256, 256, 0, stream>>>(dst, degraw, E);
    dinv_kernel<<<(n + 255) / 256, 256, 0, stream>>>(degraw, dinv, invdeg, n);

    // 3. h1 = x @ W1   (K=10 padded to 12, F=64)
    {
        int tiles = ((n + 15) / 16) * (64 / 16);
        gemm_wmma_kernel<10, 64><<<(tiles + 3) / 4, 128, 0, stream>>>(x, W1, h1, n);
    }

    // 4. agg1[dst] += h1[src]*norm  (16 float4 per edge)
    {
        long long tot = (long long)E * 16;
        edge_agg_kernel<16><<<(int)((tot + 255) / 256), 256, 0, stream>>>(
            (const float4*)h1, agg1, src, dst, dinv, E);
    }

    // 5. agg1 = relu(agg1 + h1*invdeg + b1)
    {
        long long tot = (long long)n * 64;
        post_kernel<64><<<(int)((tot + 255) / 256), 256, 0, stream>>>(agg1, h1, invdeg, b1, n);
    }

    // 6. h2 = agg1 @ W2   (K=64, F=32)
    {
        int tiles = ((n + 15) / 16) * (32 / 16);
        gemm_wmma_kernel<64, 32><<<(tiles + 3) / 4, 128, 0, stream>>>(agg1, W2, h2, n);
    }

    // 7. agg2[dst] += h2[src]*norm  (8 float4 per edge)
    {
        long long tot = (long long)E * 8;
        edge_agg_kernel<8><<<(int)((tot + 255) / 256), 256, 0, stream>>>(
            (const float4*)h2, agg2, src, dst, dinv, E);
    }

    // 8. agg2 = relu(agg2 + h2*invdeg + b2)
    {
        long long tot = (long long)n * 32;
        post_kernel<32><<<(int)((tot + 255) / 256), 256, 0, stream>>>(agg2, h2, invdeg, b2, n);
    }

    // 9. pooled[batch] += agg2
    pool_kernel<<<(n * 8 + 255) / 256, 256, 0, stream>>>((const float4*)agg2, batch, pooled, n);

    // 10. out = sigmoid(pooled @ Wfc + bfc)
    fc_kernel<<<(G + 255) / 256, 256, 0, stream>>>(pooled, Wfc, bfc, out, G);
}